// InfoNCELoss_2911987827023
// MI455X (gfx1250) — compile-verified
//
#include <hip/hip_runtime.h>
#include <math.h>

// ---------------------------------------------------------------------------
// InfoNCE loss, N=4096, D=256, fp32, tau=0.5.
// loss = mean_i [ (2+lnS1_i - s_ii) + (2+lnS2_i - s_ii) ] over 2n rows,
// using the fixed shift c = max logit = 1/tau = 2 (cosine/tau in [-2,2]).
// Gram strips computed with V_WMMA_F32_16X16X4_F32 (fp32-exact matrix path).
// Round 2: 2-way J-tile blocking -> 8 wmma per 6 loads, 8 indep acc chains.
// ---------------------------------------------------------------------------

typedef __attribute__((ext_vector_type(2))) float v2f;
typedef __attribute__((ext_vector_type(4))) float v4f;
typedef __attribute__((ext_vector_type(8))) float v8f;

#define NROWS   4096
#define DDIM    256
#define INV_TAU 2.0f
#define LOG2E   1.4426950408889634f
#define EXP_SCALE (INV_TAU * LOG2E)      // logit -> log2 domain
#define EXP_BIAS  (-INV_TAU * LOG2E)     // shift by c = INV_TAU (max logit)
#define EPS     1e-8f
#define TILE_OFF (16 * DDIM)             // +16 rows, in floats (fits IOFFSET)

#define WMMA_F32(A, B, C) \
    __builtin_amdgcn_wmma_f32_16x16x4_f32(false, (A), false, (B), (short)0, (C), false, false)

// ---------------------------------------------------------------------------
// Stage 1: one wave per row. Compute ||x||, ||y||, x.y, write xn, yn, sdiag.
// ---------------------------------------------------------------------------
__global__ __launch_bounds__(256) void nrm_kernel(const float* __restrict__ x,
                                                  const float* __restrict__ y,
                                                  float* __restrict__ xn,
                                                  float* __restrict__ yn,
                                                  float* __restrict__ sdiag) {
    const int wave = threadIdx.x >> 5;
    const int lane = threadIdx.x & 31;
    const int row  = blockIdx.x * 8 + wave;   // gridDim.x = 512

    const float* xr = x + (size_t)row * DDIM;
    const float* yr = y + (size_t)row * DDIM;

    v4f x0 = *(const v4f*)(xr + lane * 8);
    v4f x1 = *(const v4f*)(xr + lane * 8 + 4);
    v4f y0 = *(const v4f*)(yr + lane * 8);
    v4f y1 = *(const v4f*)(yr + lane * 8 + 4);

    float sxx = 0.f, syy = 0.f, sxy = 0.f;
#pragma unroll
    for (int c = 0; c < 4; ++c) {
        sxx += x0[c] * x0[c] + x1[c] * x1[c];
        syy += y0[c] * y0[c] + y1[c] * y1[c];
        sxy += x0[c] * y0[c] + x1[c] * y1[c];
    }
#pragma unroll
    for (int off = 16; off >= 1; off >>= 1) {
        sxx += __shfl_xor(sxx, off, 32);
        syy += __shfl_xor(syy, off, 32);
        sxy += __shfl_xor(sxy, off, 32);
    }
    const float invx = 1.0f / fmaxf(sqrtf(sxx), EPS);
    const float invy = 1.0f / fmaxf(sqrtf(syy), EPS);

    float* xo = xn + (size_t)row * DDIM;
    float* yo = yn + (size_t)row * DDIM;
    *(v4f*)(xo + lane * 8)     = x0 * invx;
    *(v4f*)(xo + lane * 8 + 4) = x1 * invx;
    *(v4f*)(yo + lane * 8)     = y0 * invy;
    *(v4f*)(yo + lane * 8 + 4) = y1 * invy;

    if (lane == 0) sdiag[row] = sxy * invx * invy * INV_TAU;  // logit units
}

// ---------------------------------------------------------------------------
// Stage 2: one workgroup per 16-row I-block. 8 waves sweep PAIRS of 16-col
// J-tiles. Per K-step: 6 b64 loads feed 8 WMMA f32 16x16x4.
// ---------------------------------------------------------------------------
__global__ __launch_bounds__(256) void sim_lse_kernel(const float* __restrict__ xn,
                                                      const float* __restrict__ yn,
                                                      const float* __restrict__ sdiag,
                                                      float* __restrict__ partial) {
    __shared__ float S1[16];
    __shared__ float S2[16];
    if (threadIdx.x < 16) { S1[threadIdx.x] = 0.f; S2[threadIdx.x] = 0.f; }
    __syncthreads();

    const int wave   = threadIdx.x >> 5;
    const int lane   = threadIdx.x & 31;
    const int laneLo = lane & 15;           // A: row M, B: col N, C/D: col N
    const int laneHi = lane >> 4;           // selects K pair / row half

    const int I = blockIdx.x * 16;          // gridDim.x = 256
    const float* xI = xn + (size_t)(I + laneLo) * DDIM;
    const float* yI = yn + (size_t)(I + laneLo) * DDIM;

    float rs1[8], rs2[8];
#pragma unroll
    for (int v = 0; v < 8; ++v) { rs1[v] = 0.f; rs2[v] = 0.f; }

    // 128 tile-pairs; each wave takes every 8th pair (wave-uniform => EXEC all 1s)
    for (int p = wave; p < NROWS / 32; p += 8) {
        const int J0 = p * 32;              // tiles J0 and J0+16
        const float* xJ = xn + (size_t)(J0 + laneLo) * DDIM;
        const float* yJ = yn + (size_t)(J0 + laneLo) * DDIM;

        v8f aXX0 = {}, aXY0 = {}, aYY0 = {}, aYX0 = {};
        v8f aXX1 = {}, aXY1 = {}, aYY1 = {}, aYX1 = {};
#pragma unroll 4
        for (int kk = 0; kk < DDIM / 4; ++kk) {
            const int kb = kk * 4 + laneHi * 2;       // lanes 16-31 carry K+2,K+3
            v2f ax  = *(const v2f*)(xI + kb);
            v2f ay  = *(const v2f*)(yI + kb);
            v2f bx0 = *(const v2f*)(xJ + kb);
            v2f by0 = *(const v2f*)(yJ + kb);
            v2f bx1 = *(const v2f*)(xJ + kb + TILE_OFF);   // +16 rows: imm offset
            v2f by1 = *(const v2f*)(yJ + kb + TILE_OFF);
            aXX0 = WMMA_F32(ax, bx0, aXX0);
            aXY0 = WMMA_F32(ax, by0, aXY0);
            aYY0 = WMMA_F32(ay, by0, aYY0);
            aYX0 = WMMA_F32(ay, bx0, aYX0);
            aXX1 = WMMA_F32(ax, bx1, aXX1);
            aXY1 = WMMA_F32(ax, by1, aXY1);
            aYY1 = WMMA_F32(ay, by1, aYY1);
            aYX1 = WMMA_F32(ay, bx1, aYX1);
        }

        const bool diag0 = (J0 == I);
        const bool diag1 = (J0 + 16 == I);
#pragma unroll
        for (int v = 0; v < 8; ++v) {
            const int row = laneHi * 8 + v;           // C/D: lane holds rows v(+8)
            const bool onDiag = (row == laneLo);
            const bool d0 = diag0 && onDiag;
            const bool d1 = diag1 && onDiag;
            // exp(logit - 2) via v_exp_f32 in log2 domain; logits in [-2,2].
            float e;
            e = d0 ? 0.f : exp2f(aXX0[v] * EXP_SCALE + EXP_BIAS); rs1[v] += e;
            e =            exp2f(aXY0[v] * EXP_SCALE + EXP_BIAS); rs1[v] += e;
            e = d0 ? 0.f : exp2f(aYY0[v] * EXP_SCALE + EXP_BIAS); rs2[v] += e;
            e =            exp2f(aYX0[v] * EXP_SCALE + EXP_BIAS); rs2[v] += e;
            e = d1 ? 0.f : exp2f(aXX1[v] * EXP_SCALE + EXP_BIAS); rs1[v] += e;
            e =            exp2f(aXY1[v] * EXP_SCALE + EXP_BIAS); rs1[v] += e;
            e = d1 ? 0.f : exp2f(aYY1[v] * EXP_SCALE + EXP_BIAS); rs2[v] += e;
            e =            exp2f(aYX1[v] * EXP_SCALE + EXP_BIAS); rs2[v] += e;
        }
    }

    // Column reduction: 16 lanes per half-wave hold the same 8 rows.
#pragma unroll
    for (int v = 0; v < 8; ++v) {
#pragma unroll
        for (int off = 8; off >= 1; off >>= 1) {
            rs1[v] += __shfl_xor(rs1[v], off, 32);
            rs2[v] += __shfl_xor(rs2[v], off, 32);
        }
    }
    if (laneLo == 0) {                                // lanes 0 and 16 of each wave
#pragma unroll
        for (int v = 0; v < 8; ++v) {
            const int row = laneHi * 8 + v;
            atomicAdd(&S1[row], rs1[v]);              // ds_add_f32
            atomicAdd(&S2[row], rs2[v]);
        }
    }
    __syncthreads();

    if (threadIdx.x < 16) {
        const int row = threadIdx.x;
        // (c + lnS1 - s_ii) + (c + lnS2 - s_ii), c = INV_TAU = 2
        float val = 2.0f * INV_TAU + logf(S1[row]) + logf(S2[row])
                  - 2.0f * sdiag[I + row];
#pragma unroll
        for (int off = 8; off >= 1; off >>= 1) val += __shfl_xor(val, off, 32);
        if (threadIdx.x == 0) partial[blockIdx.x] = val;
    }
}

// ---------------------------------------------------------------------------
// Stage 3: sum 256 block partials -> loss / (2N).
// ---------------------------------------------------------------------------
__global__ __launch_bounds__(256) void final_kernel(const float* __restrict__ partial,
                                                    float* __restrict__ out) {
    __shared__ float buf[8];
    const int lane = threadIdx.x & 31;
    const int wave = threadIdx.x >> 5;
    float v = partial[threadIdx.x];
#pragma unroll
    for (int off = 16; off >= 1; off >>= 1) v += __shfl_xor(v, off, 32);
    if (lane == 0) buf[wave] = v;
    __syncthreads();
    if (threadIdx.x == 0) {
        float s = 0.f;
#pragma unroll
        for (int w = 0; w < 8; ++w) s += buf[w];
        out[0] = s * (1.0f / (2.0f * NROWS));
    }
}

extern "C" void kernel_launch(void* const* d_in, const int* in_sizes, int n_in,
                              void* d_out, int out_size, void* d_ws, size_t ws_size,
                              hipStream_t stream) {
    const float* x = (const float*)d_in[0];
    const float* y = (const float*)d_in[1];

    float* ws      = (float*)d_ws;
    float* xn      = ws;                               // 4096*256
    float* yn      = xn + (size_t)NROWS * DDIM;        // 4096*256
    float* sdiag   = yn + (size_t)NROWS * DDIM;        // 4096
    float* partial = sdiag + NROWS;                    // 256

    nrm_kernel<<<NROWS / 8, 256, 0, stream>>>(x, y, xn, yn, sdiag);
    sim_lse_kernel<<<NROWS / 16, 256, 0, stream>>>(xn, yn, sdiag, partial);
    final_kernel<<<1, 256, 0, stream>>>(partial, (float*)d_out);
}